// MeanAggregator_5076651344590
// MI455X (gfx1250) — compile-verified
//
#include <hip/hip_runtime.h>

typedef __attribute__((ext_vector_type(2))) float v2f;
typedef __attribute__((ext_vector_type(8))) float v8f;

// Problem constants (from the reference): D = U = 128, K = 25 neighbours.
constexpr int Dc = 128;   // feature dim (GEMM K)
constexpr int Uc = 128;   // output dim (GEMM N)
constexpr int Kn = 25;    // neighbours per node
constexpr int TILE_M = 16;
constexpr int LDS_PITCH = 132; // 128 + 4 pad -> column reads hit 16 distinct banks

// Fused: gather 26 rows -> mean (LDS) -> 16x16 f32 WMMA GEMM -> ReLU -> out.
// One block = one 16-row M tile; 8 waves each own one 16-col N tile.
__global__ __launch_bounds__(256) void sage_mean_wmma_relu(
    const float* __restrict__ features,
    const int*   __restrict__ node,
    const int*   __restrict__ neigh,
    const float* __restrict__ W,     // [Dc x Uc] row-major
    float*       __restrict__ out,   // [B x Uc]
    int B)
{
  __shared__ float lds[TILE_M * LDS_PITCH];

  const int tile = blockIdx.x;
  const int t    = threadIdx.x;

  // ---- Phase 1: gather + mean into LDS -------------------------------
  // 256 threads cover 2 rows x 128 cols per iteration (coalesced 512B rows).
  {
    const int col = t & (Dc - 1);
    const int r0  = t >> 7;            // 0 or 1
    for (int r = r0; r < TILE_M; r += 2) {
      int gm = tile * TILE_M + r;
      gm = (gm < B) ? gm : (B - 1);    // clamp (keeps loads in-bounds; stores guarded later)
      float acc = features[(long)node[gm] * Dc + col];
      const int nb = gm * Kn;
      #pragma unroll 5
      for (int j = 0; j < Kn; ++j) {
        acc += features[(long)neigh[nb + j] * Dc + col];
      }
      lds[r * LDS_PITCH + col] = acc * (1.0f / (Kn + 1));
    }
  }
  __syncthreads();

  // ---- Phase 2: per-wave 16x16 output tile via V_WMMA_F32_16X16X4_F32 ----
  const int wave = t >> 5;            // 0..7 -> N tile index
  const int lane = t & 31;
  const int mn   = lane & 15;         // M index for A, N index for B/D
  const int hi   = lane >> 4;         // 0: K pair {0,1}, 1: K pair {2,3}
  const int kb   = hi * 2;

  const float* wcol = W + wave * 16 + mn;  // column (wave*16 + n) of W

  v8f c = {};
  #pragma unroll
  for (int kk = 0; kk < Dc; kk += 4) {
    v2f a, b;
    // A (16x4 f32): lane m holds K = kb, kb+1 of row m
    a.x = lds[mn * LDS_PITCH + kk + kb];
    a.y = lds[mn * LDS_PITCH + kk + kb + 1];
    // B (4x16 f32): lane n holds rows K = kb, kb+1 of column n
    b.x = wcol[(long)(kk + kb)     * Uc];
    b.y = wcol[(long)(kk + kb + 1) * Uc];
    c = __builtin_amdgcn_wmma_f32_16x16x4_f32(
        /*neg_a=*/false, a, /*neg_b=*/false, b,
        /*c_mod=*/(short)0, c, /*reuse_a=*/false, /*reuse_b=*/false);
  }

  // ---- ReLU + store: VGPR r -> row (r + 8*hi), col = n ----------------
  #pragma unroll
  for (int r = 0; r < 8; ++r) {
    const int row = tile * TILE_M + hi * 8 + r;
    if (row < B) {
      float v = c[r];
      out[(long)row * Uc + wave * 16 + mn] = (v > 0.0f) ? v : 0.0f;
    }
  }
}

// raw_features passthrough: float4 grid-stride copy (N*D % 4 == 0).
__global__ __launch_bounds__(256) void copy_raw_f4(
    const float4* __restrict__ src, float4* __restrict__ dst, long n4)
{
  long i = (long)blockIdx.x * blockDim.x + threadIdx.x;
  const long stride = (long)gridDim.x * blockDim.x;
  for (; i < n4; i += stride) dst[i] = src[i];
}

extern "C" void kernel_launch(void* const* d_in, const int* in_sizes, int n_in,
                              void* d_out, int out_size, void* d_ws, size_t ws_size,
                              hipStream_t stream) {
  // setup_inputs order: features, node, neighbours, raw_features, neigh_weights, i
  const float* features = (const float*)d_in[0];
  const int*   node     = (const int*)d_in[1];   // jax default x64-off -> int32
  const int*   neigh    = (const int*)d_in[2];
  const float* raw      = (const float*)d_in[3];
  const float* W        = (const float*)d_in[4];

  const int  B  = in_sizes[1];          // node is [B,1]
  const long ND = (long)in_sizes[3];    // raw_features flat count

  float* out = (float*)d_out;

  const int mtiles = (B + TILE_M - 1) / TILE_M;
  sage_mean_wmma_relu<<<mtiles, 256, 0, stream>>>(features, node, neigh, W, out, B);

  const long n4 = ND / 4;
  copy_raw_f4<<<2048, 256, 0, stream>>>((const float4*)raw,
                                        (float4*)(out + (long)B * Uc), n4);
}